// GCNLayer_352187318566
// MI455X (gfx1250) — compile-verified
//
#include <hip/hip_runtime.h>

#define N_NODES 100000
#define N_EDGES 1600000
#define NDIM_IN 64
#define EDIM 64
#define NDIM_OUT 128
#define FAN_IN (NDIM_IN + EDIM)   // 128

typedef float v2f __attribute__((ext_vector_type(2)));
typedef float v8f __attribute__((ext_vector_type(8)));

// ---------------------------------------------------------------------------
// Phase 0: zero the scatter accumulators in workspace (must run every call).
// ---------------------------------------------------------------------------
__global__ void zero_ws_kernel(float* __restrict__ agg, float* __restrict__ cnt) {
    const int stride = gridDim.x * blockDim.x;
    const int total  = N_NODES * EDIM;
    for (int i = blockIdx.x * blockDim.x + threadIdx.x; i < total; i += stride)
        agg[i] = 0.0f;
    for (int j = blockIdx.x * blockDim.x + threadIdx.x; j < N_NODES; j += stride)
        cnt[j] = 0.0f;
}

// ---------------------------------------------------------------------------
// Phase 1: per-edge scale + scatter-add (bandwidth bound, atomics land in L2).
// 16 lanes per edge; each lane handles a float4 chunk of the 64-wide feature.
// ---------------------------------------------------------------------------
__global__ void edge_scatter_kernel(const float* __restrict__ efeats,
                                    const float* __restrict__ norm_weight,
                                    const int*   __restrict__ dst,
                                    float* __restrict__ agg,
                                    float* __restrict__ cnt) {
    const long long gid = (long long)blockIdx.x * blockDim.x + threadIdx.x;
    const int e     = (int)(gid >> 4);
    const int chunk = (int)(gid & 15);
    if (e >= N_EDGES) return;

    const float w = norm_weight[e];
    const int   d = dst[e];
    const float4 v = ((const float4*)(efeats + (long long)e * EDIM))[chunk];

    float* base = agg + (long long)d * EDIM + chunk * 4;
    atomicAdd(base + 0, w * v.x);
    atomicAdd(base + 1, w * v.y);
    atomicAdd(base + 2, w * v.z);
    atomicAdd(base + 3, w * v.w);
    if (chunk == 0) atomicAdd(cnt + d, 1.0f);
}

// ---------------------------------------------------------------------------
// Phase 2: h = [nfeats | agg/max(cnt,1)]  (16 x 128 tile in LDS),
// out = relu(h @ W^T + b) via V_WMMA_F32_16X16X4_F32, K-loop of 32.
// Block = 256 threads = 8 waves; each wave owns one 16-wide output slice.
// ---------------------------------------------------------------------------
#define HS_STRIDE 130   // pad so A-fragment LDS reads spread across banks

__global__ __launch_bounds__(256)
void node_gemm_kernel(const float* __restrict__ nfeats,
                      const float* __restrict__ agg,
                      const float* __restrict__ cnt,
                      const float* __restrict__ W,      // [128 out][128 in]
                      const float* __restrict__ bias,   // [128]
                      float* __restrict__ out) {        // [N][128]
    __shared__ float hs[16 * HS_STRIDE];

    const int mbase = blockIdx.x * 16;
    const int t     = threadIdx.x;

    // Build the 16x128 concatenated-feature tile cooperatively.
#pragma unroll
    for (int i = 0; i < 8; ++i) {
        const int idx  = t + i * 256;       // 0..2047
        const int m    = idx >> 7;          // node within tile
        const int f    = idx & 127;         // feature
        const int node = mbase + m;
        float v;
        if (f < NDIM_IN) {
            v = nfeats[(long long)node * NDIM_IN + f];
        } else {
            const float c   = cnt[node];
            const float inv = 1.0f / fmaxf(c, 1.0f);
            v = agg[(long long)node * EDIM + (f - NDIM_IN)] * inv;
        }
        hs[m * HS_STRIDE + f] = v;
    }
    __syncthreads();

    const int wave  = t >> 5;        // 0..7 -> output-channel slice
    const int lane  = t & 31;
    const int nbase = wave * 16;
    const int n     = lane & 15;     // output channel within slice
    const int half  = lane >> 4;     // 0/1 (K-half selector per ISA layout)
    const int arow  = lane & 15;     // A-matrix row (M)

    v8f c = {};
#pragma unroll
    for (int k0 = 0; k0 < FAN_IN; k0 += 4) {
        const int ka = k0 + half * 2;
        // A fragment: lanes 0-15 hold K={k0,k0+1}, lanes 16-31 K={k0+2,k0+3}
        v2f a;
        a.x = hs[arow * HS_STRIDE + ka];
        a.y = hs[arow * HS_STRIDE + ka + 1];
        // B fragment: B[k][n] = W[n][k]; VGPR0->K=half*2, VGPR1->K=half*2+1
        v2f bm;
        bm.x = W[(nbase + n) * FAN_IN + ka];
        bm.y = W[(nbase + n) * FAN_IN + ka + 1];
        c = __builtin_amdgcn_wmma_f32_16x16x4_f32(
                /*neg_a=*/false, a, /*neg_b=*/false, bm,
                /*c_mod=*/(short)0, c, /*reuse_a=*/false, /*reuse_b=*/false);
    }

    const float bb = bias[nbase + n];
    // C/D layout: VGPR i -> row (i + half*8), col n.
#pragma unroll
    for (int i = 0; i < 8; ++i) {
        const int   mr = i + half * 8;
        const float v  = fmaxf(c[i] + bb, 0.0f);
        out[(long long)(mbase + mr) * NDIM_OUT + nbase + n] = v;
    }
}

// ---------------------------------------------------------------------------
// Host-side launcher (graph-capture safe: only kernel launches on `stream`).
// ---------------------------------------------------------------------------
extern "C" void kernel_launch(void* const* d_in, const int* in_sizes, int n_in,
                              void* d_out, int out_size, void* d_ws, size_t ws_size,
                              hipStream_t stream) {
    const float* nfeats      = (const float*)d_in[0];  // [N,1,64]
    const float* efeats      = (const float*)d_in[1];  // [E,1,64]
    const float* norm_weight = (const float*)d_in[2];  // [E]
    const int*   dst         = (const int*)  d_in[3];  // [E]
    const float* W           = (const float*)d_in[4];  // [128,128]
    const float* bias        = (const float*)d_in[5];  // [128]
    float*       out         = (float*)d_out;          // [N,128]

    float* agg = (float*)d_ws;                         // N*64 floats
    float* cnt = agg + (size_t)N_NODES * EDIM;         // N floats

    zero_ws_kernel<<<1024, 256, 0, stream>>>(agg, cnt);

    const long long scatter_threads = (long long)N_EDGES * 16;
    const int scatter_blocks = (int)((scatter_threads + 255) / 256);
    edge_scatter_kernel<<<scatter_blocks, 256, 0, stream>>>(
        efeats, norm_weight, dst, agg, cnt);

    node_gemm_kernel<<<N_NODES / 16, 256, 0, stream>>>(
        nfeats, agg, cnt, W, bias, out);
}